// GNN_18545668784680
// MI455X (gfx1250) — compile-verified
//
#include <hip/hip_runtime.h>

typedef __attribute__((ext_vector_type(2))) float v2f;
typedef __attribute__((ext_vector_type(8))) float v8f;

#define D 64

// ---------------------------------------------------------------------------
// Zero-fill workspace buffer (grid-stride, float4 stores).
// n4 = element count / 4.
// ---------------------------------------------------------------------------
__global__ void gnn_zero(float4* __restrict__ p, long n4) {
  long i = (long)blockIdx.x * blockDim.x + threadIdx.x;
  long stride = (long)gridDim.x * blockDim.x;
  float4 z = make_float4(0.f, 0.f, 0.f, 0.f);
  for (; i < n4; i += stride) p[i] = z;
}

// ---------------------------------------------------------------------------
// Edge scatter-add: one wave per edge, each lane owns a float2 slice of the
// 64-wide feature row. Edge id is forced scalar (readfirstlane) so the two
// index fetches become s_load_b32 instead of 32 redundant vector loads; the
// feature gather is one coalesced 256B global_load_b64 per wave, followed by
// non-returning global_atomic_add_f32 into the L2-resident AGG buffer.
// ---------------------------------------------------------------------------
__global__ void gnn_scatter(const float* __restrict__ X,
                            const int* __restrict__ src,
                            const int* __restrict__ dst,
                            float* __restrict__ AGG,
                            int nEdges) {
  long gid = (long)blockIdx.x * blockDim.x + threadIdx.x;
  int edge = (int)(gid >> 5);
  int lane = (int)(gid & 31);
  if (edge >= nEdges) return;              // wave-uniform: EXEC stays full
  edge = __builtin_amdgcn_readfirstlane(edge);
  int s = src[edge];                       // scalar loads (uniform address)
  int d = dst[edge];
  v2f v = *(const v2f*)(X + (long)s * D + lane * 2);
  float* out = AGG + (long)d * D + lane * 2;
  atomicAdd(out + 0, v.x);
  atomicAdd(out + 1, v.y);
}

// ---------------------------------------------------------------------------
// Fused GraphConv GEMM epilogue:
//   OUT[16-node tile] = act( AGG @ Wrel^T + bias + X @ Wroot^T )
// 256 threads = 8 waves per block; both 64x64 weight matrices are staged in
// LDS once per block (32KB), then each wave computes a 16x64 output tile with
// f32 WMMA 16x16x4, pulling B fragments from LDS (ds_load_b64) and A
// fragments from global (streamed node rows).
//
// A fragment (16x4 f32, 2 VGPRs): lane L holds row M=L&15, K = {k0, k0+1},
//   k0 = 4*ks + 2*(L>>4).
// B fragment (4x16 f32, 2 VGPRs): lane L holds col N=L&15, same K pair ->
//   row-major float2 of W[o][k] at o = nt*16 + (L&15), k = k0 (i.e. W^T).
// C/D (16x16 f32, 8 VGPRs): lane L, VGPR r -> (M = r + 8*(L>>4), N = L&15).
// ---------------------------------------------------------------------------
__global__ void __launch_bounds__(256)
gnn_gemm(const float* __restrict__ AGG,
         const float* __restrict__ X,
         const float* __restrict__ Wrel,
         const float* __restrict__ Wroot,
         const float* __restrict__ bias,
         float* __restrict__ OUT,
         int nNodes, int doRelu) {
  __shared__ float sW[2][D * D];           // 32 KB: [0]=Wrel, [1]=Wroot

  const int tid = threadIdx.x;

  // Cooperative weight stage: 2 * 4096 floats with float4 per thread.
  {
    const float4* g0 = (const float4*)Wrel;
    const float4* g1 = (const float4*)Wroot;
    float4* s0 = (float4*)sW[0];
    float4* s1 = (float4*)sW[1];
#pragma unroll
    for (int i = 0; i < (D * D / 4) / 256; ++i) {
      int idx = tid + i * 256;
      s0[idx] = g0[idx];
      s1[idx] = g1[idx];
    }
  }
  __syncthreads();

  const int lane  = tid & 31;
  const int wave  = tid >> 5;
  const int row   = lane & 15;
  const int khalf = lane >> 4;
  const long nodeBase = ((long)blockIdx.x * 8 + wave) * 16;

  // Clamp load row so EXEC stays all-1s (WMMA requirement); stores are guarded.
  long ldRow = nodeBase + row;
  if (ldRow >= nNodes) ldRow = nNodes - 1;
  const float* arow = AGG + ldRow * D;
  const float* xrow = X   + ldRow * D;

  v8f acc[4];
#pragma unroll
  for (int nt = 0; nt < 4; ++nt) {
    float bv = bias[nt * 16 + row];        // bias depends only on N
#pragma unroll
    for (int r = 0; r < 8; ++r) acc[nt][r] = bv;
  }

#pragma unroll 4
  for (int ks = 0; ks < 16; ++ks) {        // K = 64 in steps of 4
    const int k0 = ks * 4 + khalf * 2;
    v2f aAgg = *(const v2f*)(arow + k0);
    v2f aX   = *(const v2f*)(xrow + k0);
#pragma unroll
    for (int nt = 0; nt < 4; ++nt) {
      const int n = nt * 16 + row;
      v2f bRel  = *(const v2f*)(&sW[0][n * D + k0]);   // ds_load_b64
      v2f bRoot = *(const v2f*)(&sW[1][n * D + k0]);
      acc[nt] = __builtin_amdgcn_wmma_f32_16x16x4_f32(
          false, aAgg, false, bRel, (short)0, acc[nt], false, false);
      acc[nt] = __builtin_amdgcn_wmma_f32_16x16x4_f32(
          false, aX,   false, bRoot, (short)0, acc[nt], false, false);
    }
  }

#pragma unroll
  for (int nt = 0; nt < 4; ++nt) {
    const int n = nt * 16 + row;
#pragma unroll
    for (int r = 0; r < 8; ++r) {
      long m = nodeBase + khalf * 8 + r;
      if (m < nNodes) {
        float v = acc[nt][r];
        if (doRelu) v = fmaxf(v, 0.0f);
        OUT[m * D + n] = v;
      }
    }
  }
}

// ---------------------------------------------------------------------------
// Launch: two GraphConv layers.
//   layer1: agg = scatter(x); h   = relu(agg@Wrel1^T + b1 + x@Wroot1^T)
//   layer2: agg = scatter(h); out =      agg@Wrel2^T + b2 + h@Wroot2^T
// ---------------------------------------------------------------------------
extern "C" void kernel_launch(void* const* d_in, const int* in_sizes, int n_in,
                              void* d_out, int out_size, void* d_ws, size_t ws_size,
                              hipStream_t stream) {
  const float* x      = (const float*)d_in[0];
  const int*   ei     = (const int*)  d_in[1];
  const float* Wrel1  = (const float*)d_in[2];
  const float* brel1  = (const float*)d_in[3];
  const float* Wroot1 = (const float*)d_in[4];
  const float* Wrel2  = (const float*)d_in[5];
  const float* brel2  = (const float*)d_in[6];
  const float* Wroot2 = (const float*)d_in[7];

  const int nNodes = in_sizes[0] / D;
  const int nEdges = in_sizes[1] / 2;
  const int* src = ei;            // edge_index[0]
  const int* dst = ei + nEdges;   // edge_index[1]

  float* agg = (float*)d_ws;                       // N*D f32
  float* h   = agg + (size_t)nNodes * D;           // N*D f32
  float* out = (float*)d_out;

  const long nElems4 = (long)nNodes * D / 4;
  const int  zeroBlocks    = 2048;
  const int  tiles         = (nNodes + 15) / 16;
  const int  gemmBlocks    = (tiles + 7) / 8;      // 8 waves (tiles) per block
  const long scatterThread = (long)nEdges * 32;
  const int  scatterBlocks = (int)((scatterThread + 255) / 256);

  // ---- layer 1 ----
  gnn_zero   <<<zeroBlocks,    256, 0, stream>>>((float4*)agg, nElems4);
  gnn_scatter<<<scatterBlocks, 256, 0, stream>>>(x, src, dst, agg, nEdges);
  gnn_gemm   <<<gemmBlocks,    256, 0, stream>>>(agg, x, Wrel1, Wroot1, brel1, h, nNodes, 1);

  // ---- layer 2 ----
  gnn_zero   <<<zeroBlocks,    256, 0, stream>>>((float4*)agg, nElems4);
  gnn_scatter<<<scatterBlocks, 256, 0, stream>>>(h, src, dst, agg, nEdges);
  gnn_gemm   <<<gemmBlocks,    256, 0, stream>>>(agg, h, Wrel2, Wroot2, brel2, out, nNodes, 0);
}